// ImageTransformer2D_3D_MS_2645699854386
// MI455X (gfx1250) — compile-verified
//
#include <hip/hip_runtime.h>
#include <hip/hip_bf16.h>
#include <math.h>

// ---------------- problem constants ----------------
#define Bq   2
#define Vv   6
#define Cc   128
#define NQn  1200
#define Hh   8
#define PTSn 4
#define FFNn 256
#define LVLS 4
#define LTOT 29750
#define APAD 4   // LDS row padding (floats) to spread banks

typedef __attribute__((ext_vector_type(2))) float v2f;
typedef __attribute__((ext_vector_type(8))) float v8f;

__device__ __forceinline__ float sigm(float x) { return 1.0f / (1.0f + expf(-x)); }

// ---------------------------------------------------------------------------
// Wave-level fp32 WMMA 16x16 tile:  acc += A(16xK, LDS) * W(KxN, global).
// K, N compile-time: fully unrolled chain of v_wmma_f32_16x16x4_f32 with
// immediate-offset global loads (no per-iteration address math, no guards).
// VGPR layout per CDNA5 ISA 7.12.2:
//   A 16x4 f32: lanes 0-15 M=0..15 {VGPR0:K=k, VGPR1:K=k+1};
//               lanes 16-31 M=0..15 {VGPR0:K=k+2, VGPR1:K=k+3}
//   B 4x16 f32: mirrored; D 16x16: VGPR r -> rows r / r+8
// ---------------------------------------------------------------------------
template<int K, int N, int LDA>
__device__ __forceinline__ v8f wmma_tile(const float* __restrict__ As,
                                         const float* __restrict__ W,
                                         int col0, v8f acc) {
  const int lane = threadIdx.x & 31;
  const int half = lane >> 4;
  const int ln   = lane & 15;
  const float* Ap = As + ln * LDA + 2 * half;
  const float* Wp = W + (size_t)(2 * half) * N + col0 + ln;
  __builtin_prefetch(Wp, 0, 3);
#pragma unroll
  for (int k = 0; k < K; k += 4) {
    v2f a, b;
    a.x = Ap[k];
    a.y = Ap[k + 1];
    b.x = Wp[(size_t)k * N];
    b.y = Wp[(size_t)k * N + N];
    acc = __builtin_amdgcn_wmma_f32_16x16x4_f32(false, a, false, b, (short)0, acc,
                                                false, false);
  }
  return acc;
}

template<int N>
__device__ __forceinline__ void store_tile(float* __restrict__ out, int row0,
                                           int nrows, int col0, v8f acc,
                                           const float* __restrict__ bias,
                                           bool relu) {
  const int lane = threadIdx.x & 31;
  const int half = lane >> 4;
  const int ln   = lane & 15;
  const int col  = col0 + ln;
  const float bv = bias ? bias[col] : 0.0f;
#pragma unroll
  for (int r = 0; r < 8; ++r) {
    int row = row0 + r + 8 * half;
    if (row < nrows) {
      float v = acc[r] + bv;
      if (relu) v = fmaxf(v, 0.0f);
      out[(size_t)row * N + col] = v;
    }
  }
}

// ---------------------------------------------------------------------------
// Generic GEMM: out[M,N] = act(A[M,K] @ W[K,N] + bias).  256 thr = 8 waves,
// 16 rows per block; waves loop over 16-col tiles.  N multiple of 16.
// ---------------------------------------------------------------------------
template<int K, int N>
__global__ __launch_bounds__(256) void gemm_kernel(const float* __restrict__ A,
                                                   const float* __restrict__ W,
                                                   const float* __restrict__ bias,
                                                   float* __restrict__ out,
                                                   int M, int relu) {
  constexpr int LDA = K + APAD;
  __shared__ float As[16 * LDA];
  const int row0 = blockIdx.x * 16;
  for (int i = threadIdx.x; i < 16 * K; i += 256) {
    int r = i / K, c = i - r * K;
    int row = row0 + r; if (row >= M) row = M - 1;
    As[r * LDA + c] = A[(size_t)row * K + c];
  }
  __syncthreads();
  const int wave = threadIdx.x >> 5;
#pragma unroll 1
  for (int ct = wave; ct < (N >> 4); ct += 8) {
    v8f acc = {};
    acc = wmma_tile<K, N, LDA>(As, W, ct * 16, acc);
    store_tile<N>(out, row0, M, ct * 16, acc, bias, relu != 0);
  }
}

// ---------------------------------------------------------------------------
// 2->128 MLP (pos embed): out = relu(pos@w1+b1)@w2+b2 ; w1[2,128], w2[128,128]
// ---------------------------------------------------------------------------
__global__ __launch_bounds__(256) void mlp2_kernel(const float* __restrict__ pos,
                                                   const float* __restrict__ w1,
                                                   const float* __restrict__ b1,
                                                   const float* __restrict__ w2,
                                                   const float* __restrict__ b2,
                                                   float* __restrict__ out, int M) {
  constexpr int LDA = Cc + APAD;
  __shared__ float Hs[16 * LDA];
  const int row0 = blockIdx.x * 16;
  for (int i = threadIdx.x; i < 16 * Cc; i += 256) {
    int r = i >> 7, c = i & 127;
    int row = row0 + r; if (row >= M) row = M - 1;
    float h = fmaf(pos[row * 2 + 0], w1[c], fmaf(pos[row * 2 + 1], w1[Cc + c], b1[c]));
    Hs[r * LDA + c] = fmaxf(h, 0.0f);
  }
  __syncthreads();
  const int wave = threadIdx.x >> 5;  // 8 waves x 16 cols = 128
  v8f acc = {};
  acc = wmma_tile<Cc, Cc, LDA>(Hs, w2, wave * 16, acc);
  store_tile<Cc>(out, row0, M, wave * 16, acc, b2, false);
}

// ---------------------------------------------------------------------------
// value[bv, start+l, d] = sum_c feat[bv, c, l] * W[c,d] + kpeW[start+l, d]
// (kpeW already includes d_val_b).  A is transposed-strided in global; the
// WMMA accumulator is seeded from the kpeW tile (fused bias + pos-embed).
// ---------------------------------------------------------------------------
__global__ __launch_bounds__(256) void value_kernel(const float* __restrict__ feat,
                                                    int lsize, int start,
                                                    const float* __restrict__ W,
                                                    const float* __restrict__ kpeW,
                                                    float* __restrict__ value) {
  constexpr int LDA = Cc + APAD;
  __shared__ float As[16 * LDA];
  const int bv = blockIdx.y;
  const int l0 = blockIdx.x * 16;
  for (int i = threadIdx.x; i < 16 * Cc; i += 256) {
    int r = i & 15, c = i >> 4;               // consecutive tids -> consecutive l
    int idx = l0 + r; if (idx >= lsize) idx = lsize - 1;
    As[r * LDA + c] = feat[((size_t)bv * Cc + c) * lsize + idx];
  }
  __syncthreads();
  const int wave = threadIdx.x >> 5;
  const int lane = threadIdx.x & 31;
  const int half = lane >> 4, ln = lane & 15;
  const int col  = wave * 16 + ln;
  v8f acc;
#pragma unroll
  for (int r = 0; r < 8; ++r) {
    int idx = l0 + r + 8 * half; if (idx >= lsize) idx = lsize - 1;
    acc[r] = kpeW[(size_t)(start + idx) * Cc + col];
  }
  acc = wmma_tile<Cc, Cc, LDA>(As, W, wave * 16, acc);
#pragma unroll
  for (int r = 0; r < 8; ++r) {
    int idx = l0 + r + 8 * half;
    if (idx < lsize)
      value[((size_t)bv * LTOT + start + idx) * Cc + col] = acc[r];
  }
}

// ------------------------- elementwise helpers -----------------------------
__global__ void transpose_in_kernel(const float* __restrict__ iqf,
                                    float* __restrict__ x0) {
  int i = blockIdx.x * blockDim.x + threadIdx.x;
  if (i >= Bq * NQn * Cc) return;
  int c = i & 127; int n = (i >> 7) % NQn; int b = i / (NQn * Cc);
  x0[i] = iqf[((size_t)b * Cc + c) * NQn + n];
}

__global__ void transpose_out_kernel(const float* __restrict__ x,
                                     float* __restrict__ o) {
  int i = blockIdx.x * blockDim.x + threadIdx.x;
  if (i >= Bq * NQn * Cc) return;
  int n = i % NQn; int c = (i / NQn) & 127; int b = i / (NQn * Cc);
  o[i] = x[((size_t)b * NQn + n) * Cc + c];
}

__global__ void add_kernel(const float* __restrict__ a, const float* __restrict__ b,
                           float* __restrict__ o, int n) {
  int i = blockIdx.x * blockDim.x + threadIdx.x;
  if (i < n) o[i] = a[i] + b[i];
}

__global__ void counts_kernel(const int* __restrict__ view, int* __restrict__ counts) {
  __shared__ int c[Bq * Vv];
  if (threadIdx.x < Bq * Vv) c[threadIdx.x] = 0;
  __syncthreads();
  for (int i = threadIdx.x; i < Bq * NQn; i += blockDim.x) {
    int b = i / NQn;
    atomicAdd(&c[b * Vv + view[i]], 1);
  }
  __syncthreads();
  if (threadIdx.x < Bq * Vv) counts[threadIdx.x] = c[threadIdx.x];
}

// pad ph_w2[128,6] -> [128,16], ph_b2[6] -> [16] (zero fill)
__global__ void padw6_kernel(const float* __restrict__ w, const float* __restrict__ b,
                             float* __restrict__ wp, float* __restrict__ bp) {
  int i = blockIdx.x * blockDim.x + threadIdx.x;
  if (i < Cc * 16) {
    int r = i >> 4, c = i & 15;
    wp[i] = (c < 6) ? w[r * 6 + c] : 0.0f;
  }
  if (i < 16) bp[i] = (i < 6) ? b[i] : 0.0f;
}

// --------------------- view-masked self attention --------------------------
// one wave per (b, h, query); 8 queries per 256-thread block
__global__ __launch_bounds__(256) void attn_kernel(const float* __restrict__ qkv,
                                                   const int* __restrict__ view,
                                                   float* __restrict__ outb) {
  const int b = blockIdx.z, h = blockIdx.y;
  const int wave = threadIdx.x >> 5, lane = threadIdx.x & 31;
  const int n = blockIdx.x * 8 + wave;
  const float* qrow = qkv + ((size_t)(b * NQn + n) * 3 * Cc) + h * 16;
  float q[16];
#pragma unroll
  for (int d = 0; d < 16; ++d) q[d] = qrow[d] * 0.25f;  // 1/sqrt(16)
  const int vq = view[b * NQn + n];
  float mx = -3.0e38f;
  for (int m = lane; m < NQn; m += 32) {
    const float* krow = qkv + (size_t)(b * NQn + m) * 3 * Cc + Cc + h * 16;
    float dt = 0.0f;
#pragma unroll
    for (int d = 0; d < 16; ++d) dt += q[d] * krow[d];
    float lg = (view[b * NQn + m] == vq) ? dt : -1.0e9f;
    mx = fmaxf(mx, lg);
  }
#pragma unroll
  for (int s = 16; s; s >>= 1) mx = fmaxf(mx, __shfl_xor(mx, s, 32));
  float sum = 0.0f, acc[16];
#pragma unroll
  for (int d = 0; d < 16; ++d) acc[d] = 0.0f;
  for (int m = lane; m < NQn; m += 32) {
    const float* krow = qkv + (size_t)(b * NQn + m) * 3 * Cc + Cc + h * 16;
    float dt = 0.0f;
#pragma unroll
    for (int d = 0; d < 16; ++d) dt += q[d] * krow[d];
    float lg = (view[b * NQn + m] == vq) ? dt : -1.0e9f;
    float e = expf(lg - mx);
    sum += e;
    const float* vrow = qkv + (size_t)(b * NQn + m) * 3 * Cc + 2 * Cc + h * 16;
#pragma unroll
    for (int d = 0; d < 16; ++d) acc[d] += e * vrow[d];
  }
#pragma unroll
  for (int s = 16; s; s >>= 1) sum += __shfl_xor(sum, s, 32);
#pragma unroll
  for (int d = 0; d < 16; ++d) {
#pragma unroll
    for (int s = 16; s; s >>= 1) acc[d] += __shfl_xor(acc[d], s, 32);
  }
  if (lane == 0) {
    float inv = 1.0f / sum;
    float* orow = outb + (size_t)(b * NQn + n) * Cc + h * 16;
#pragma unroll
    for (int d = 0; d < 16; ++d) orow[d] = acc[d] * inv;
  }
}

// ----------------------- LayerNorm (+residual, +mask) ----------------------
__global__ __launch_bounds__(256) void ln_kernel(const float* __restrict__ a,
                                                 const float* __restrict__ res,
                                                 const float* __restrict__ g,
                                                 const float* __restrict__ beta,
                                                 float* __restrict__ out, int M,
                                                 const int* __restrict__ view,
                                                 const int* __restrict__ counts) {
  const int row = blockIdx.x * 8 + (threadIdx.x >> 5);
  if (row >= M) return;
  const int lane = threadIdx.x & 31;
  const float* ar = a   + (size_t)row * Cc;
  const float* rr = res + (size_t)row * Cc;
  float x[4]; float s = 0.0f;
#pragma unroll
  for (int j = 0; j < 4; ++j) { int c = lane * 4 + j; x[j] = ar[c] + rr[c]; s += x[j]; }
#pragma unroll
  for (int t = 16; t; t >>= 1) s += __shfl_xor(s, t, 32);
  const float mean = s * (1.0f / Cc);
  float vs = 0.0f;
#pragma unroll
  for (int j = 0; j < 4; ++j) { float d = x[j] - mean; vs += d * d; }
#pragma unroll
  for (int t = 16; t; t >>= 1) vs += __shfl_xor(vs, t, 32);
  const float rstd = rsqrtf(vs * (1.0f / Cc) + 1e-5f);
  float msk = 1.0f;
  if (view) {
    int b = row / NQn;
    msk = (counts[b * Vv + view[row]] > 1) ? 1.0f : 0.0f;
  }
#pragma unroll
  for (int j = 0; j < 4; ++j) {
    int c = lane * 4 + j;
    out[(size_t)row * Cc + c] = ((x[j] - mean) * rstd * g[c] + beta[c]) * msk;
  }
}

// ------------------- per-head softmax over LVLS*PTS = 16 -------------------
__global__ void softmax16_kernel(float* __restrict__ aw, int total) {
  int i = blockIdx.x * blockDim.x + threadIdx.x;
  if (i >= total) return;
  float* p = aw + (size_t)i * 16;
  float mx = p[0];
#pragma unroll
  for (int j = 1; j < 16; ++j) mx = fmaxf(mx, p[j]);
  float e[16], s = 0.0f;
#pragma unroll
  for (int j = 0; j < 16; ++j) { e[j] = expf(p[j] - mx); s += e[j]; }
  float inv = 1.0f / s;
#pragma unroll
  for (int j = 0; j < 16; ++j) p[j] = e[j] * inv;
}

// ------------- multi-scale deformable sampling (bilinear gather) -----------
// block = 128 threads per (b,n): thread -> (h = t/16, d = t%16)
__global__ __launch_bounds__(128) void deform_kernel(const float* __restrict__ value,
                                                     const float* __restrict__ offb,
                                                     const float* __restrict__ aw,
                                                     const float* __restrict__ qpos,
                                                     const int* __restrict__ view,
                                                     float* __restrict__ samp) {
  const int SHW[4] = {200, 100, 50, 25};
  const int SHH[4] = {112, 56, 28, 14};
  const int SST[4] = {0, 22400, 28000, 29400};
  const int b = blockIdx.y, n = blockIdx.x;
  const int t = threadIdx.x, h = t >> 4, d = t & 15;
  const int vw = view[b * NQn + n];
  const size_t vbase = (size_t)(b * Vv + vw) * LTOT;
  const float rx = sigm(qpos[(b * NQn + n) * 2 + 0]);
  const float ry = sigm(qpos[(b * NQn + n) * 2 + 1]);
  const float* orow = offb + (size_t)(b * NQn + n) * 256;
  const float* arow = aw   + (size_t)(b * NQn + n) * Cc;
  float acc = 0.0f;
#pragma unroll
  for (int l = 0; l < LVLS; ++l) {
    const int wdt = SHW[l], hgt = SHH[l], st = SST[l];
#pragma unroll
    for (int p = 0; p < PTSn; ++p) {
      const int oi = (((h * LVLS) + l) * PTSn + p) * 2;
      const float lx = (rx + orow[oi]     / (float)wdt) * (float)wdt - 0.5f;
      const float ly = (ry + orow[oi + 1] / (float)hgt) * (float)hgt - 0.5f;
      const float x0f = floorf(lx), y0f = floorf(ly);
      const float fx = lx - x0f, fy = ly - y0f;
      const int x0 = (int)x0f, y0 = (int)y0f;
      const float a = arow[h * 16 + l * PTSn + p];
      float sv = 0.0f;
#pragma unroll
      for (int dy = 0; dy < 2; ++dy) {
#pragma unroll
        for (int dx = 0; dx < 2; ++dx) {
          int xi = x0 + dx, yi = y0 + dy;
          if (xi < 0 || xi >= wdt || yi < 0 || yi >= hgt) continue;
          float wgt = (dx ? fx : 1.0f - fx) * (dy ? fy : 1.0f - fy);
          size_t rowi = vbase + st + (size_t)yi * wdt + xi;
          sv += wgt * value[rowi * Cc + h * 16 + d];
        }
      }
      acc += a * sv;
    }
  }
  samp[(size_t)(b * NQn + n) * Cc + h * 16 + d] = acc;
}

// --------------------------- final head assembly ---------------------------
__global__ void final_kernel(const float* __restrict__ res6,
                             const float* __restrict__ qpos,
                             float* __restrict__ o_newpos, float* __restrict__ o_center,
                             float* __restrict__ o_offs, float* __restrict__ o_bbox) {
  int i = blockIdx.x * blockDim.x + threadIdx.x;
  if (i >= Bq * NQn) return;
  const int b = i / NQn, n = i % NQn;
  const float* r = res6 + (size_t)i * 16;   // padded row stride
  const float crx = r[0] + qpos[i * 2 + 0];
  const float cry = r[1] + qpos[i * 2 + 1];
  o_newpos[i * 2 + 0] = crx;
  o_newpos[i * 2 + 1] = cry;
  const float c0 = sigm(crx), c1 = sigm(cry);
  o_center[(size_t)b * 2 * NQn + n]       = c0;
  o_center[(size_t)b * 2 * NQn + NQn + n] = c1;
  const float o0 = sigm(r[2]), o1 = sigm(r[3]), o2 = sigm(r[4]), o3 = sigm(r[5]);
  float* ob = o_offs + (size_t)b * 4 * NQn;
  ob[n] = o0; ob[NQn + n] = o1; ob[2 * NQn + n] = o2; ob[3 * NQn + n] = o3;
  float* bb = o_bbox + (size_t)b * 4 * NQn;
  bb[n]           = (2.0f * c0 - o0 + o2) * 0.5f;
  bb[NQn + n]     = (2.0f * c1 - o1 + o3) * 0.5f;
  bb[2 * NQn + n] = o0 + o2;
  bb[3 * NQn + n] = o1 + o3;
}

// ===========================================================================
extern "C" void kernel_launch(void* const* d_in, const int* in_sizes, int n_in,
                              void* d_out, int out_size, void* d_ws, size_t ws_size,
                              hipStream_t stream) {
  const float* iqf     = (const float*)d_in[0];   // [B,C,NQ]
  const float* qpos    = (const float*)d_in[1];   // [B,NQ,2]
  const float* feats[4]= {(const float*)d_in[2], (const float*)d_in[3],
                          (const float*)d_in[4], (const float*)d_in[5]};
  const float* kstack  = (const float*)d_in[6];   // [1,LTOT,2]
  const float* sp_w1 = (const float*)d_in[7];  const float* sp_b1 = (const float*)d_in[8];
  const float* sp_w2 = (const float*)d_in[9];  const float* sp_b2 = (const float*)d_in[10];
  const float* cp_w1 = (const float*)d_in[11]; const float* cp_b1 = (const float*)d_in[12];
  const float* cp_w2 = (const float*)d_in[13]; const float* cp_b2 = (const float*)d_in[14];
  const float* sa_wqkv = (const float*)d_in[15]; const float* sa_bqkv = (const float*)d_in[16];
  const float* sa_wo   = (const float*)d_in[17]; const float* sa_bo   = (const float*)d_in[18];
  const float* ln1_g = (const float*)d_in[19]; const float* ln1_b = (const float*)d_in[20];
  const float* d_val_w = (const float*)d_in[21]; const float* d_val_b = (const float*)d_in[22];
  const float* d_off_w = (const float*)d_in[23]; const float* d_off_b = (const float*)d_in[24];
  const float* d_attn_w= (const float*)d_in[25]; const float* d_attn_b= (const float*)d_in[26];
  const float* d_out_w = (const float*)d_in[27]; const float* d_out_b = (const float*)d_in[28];
  const float* ln2_g = (const float*)d_in[29]; const float* ln2_b = (const float*)d_in[30];
  const float* ffn_w1 = (const float*)d_in[31]; const float* ffn_b1 = (const float*)d_in[32];
  const float* ffn_w2 = (const float*)d_in[33]; const float* ffn_b2 = (const float*)d_in[34];
  const float* ln3_g = (const float*)d_in[35]; const float* ln3_b = (const float*)d_in[36];
  const float* ph_w1 = (const float*)d_in[37]; const float* ph_b1 = (const float*)d_in[38];
  const float* ph_w2 = (const float*)d_in[39]; const float* ph_b2 = (const float*)d_in[40];
  const int*   view  = (const int*)d_in[41];

  const int M = Bq * NQn;           // 2400 token rows
  const int ROWELEMS = M * Cc;      // 307200

  // ---- workspace carve-up (floats) ----
  float* ws = (float*)d_ws;
  size_t o = 0;
  auto alloc = [&](size_t n) { float* p = ws + o; o += (n + 15) & ~(size_t)15; return p; };
  float* qpe    = alloc(ROWELEMS);
  float* kpe    = alloc((size_t)LTOT * Cc);
  float* kpeW   = alloc((size_t)LTOT * Cc);
  float* value  = alloc((size_t)Bq * Vv * LTOT * Cc);
  float* x0     = alloc(ROWELEMS);
  float* qc     = alloc(ROWELEMS);
  float* qkv    = alloc((size_t)M * 3 * Cc);
  float* attnO  = alloc(ROWELEMS);
  float* tmp    = alloc(ROWELEMS);
  float* x1     = alloc(ROWELEMS);
  float* offb   = alloc((size_t)M * 256);
  float* awb    = alloc(ROWELEMS);
  float* samp   = alloc(ROWELEMS);
  float* x2     = alloc(ROWELEMS);
  float* ffnh   = alloc((size_t)M * FFNn);
  float* x3     = alloc(ROWELEMS);
  float* phh    = alloc(ROWELEMS);
  float* res6   = alloc((size_t)M * 16);   // padded to 16 cols
  float* w2pad  = alloc(Cc * 16);
  float* b2pad  = alloc(16);
  int*   counts = (int*)alloc(16);

  // ---- output carve-up ----
  float* o_x      = (float*)d_out;          // [B,C,NQ]
  float* o_newpos = o_x + ROWELEMS;         // [B,NQ,2]
  float* o_center = o_newpos + M * 2;       // [B,2,NQ]
  float* o_offs   = o_center + M * 2;       // [B,4,NQ]
  float* o_bbox   = o_offs + M * 4;         // [B,4,NQ]

  const int rowTiles = (M + 15) / 16;          // 150
  const int kpeTiles = (LTOT + 15) / 16;       // 1860

  // 1) x0 = transpose(img_query_feat); pad ph head weights
  transpose_in_kernel<<<(ROWELEMS + 255) / 256, 256, 0, stream>>>(iqf, x0);
  padw6_kernel<<<8, 256, 0, stream>>>(ph_w2, ph_b2, w2pad, b2pad);
  // 2) qpe / kpe position-embed MLPs (WMMA second layer)
  mlp2_kernel<<<rowTiles, 256, 0, stream>>>(qpos, sp_w1, sp_b1, sp_w2, sp_b2, qpe, M);
  mlp2_kernel<<<kpeTiles, 256, 0, stream>>>(kstack, cp_w1, cp_b1, cp_w2, cp_b2, kpe, LTOT);
  // 3) kpeW = kpe @ d_val_w + d_val_b
  gemm_kernel<Cc, Cc><<<kpeTiles, 256, 0, stream>>>(kpe, d_val_w, d_val_b, kpeW, LTOT, 0);
  // 4) value projection per level (dominant GEMM, 11.7 GFLOP fp32 WMMA)
  {
    const int lsz[4] = {112 * 200, 56 * 100, 28 * 50, 14 * 25};
    const int sst[4] = {0, 22400, 28000, 29400};
    for (int l = 0; l < 4; ++l) {
      dim3 grid((lsz[l] + 15) / 16, Bq * Vv);
      value_kernel<<<grid, 256, 0, stream>>>(feats[l], lsz[l], sst[l],
                                             d_val_w, kpeW, value);
    }
  }
  // 5) self-attention branch
  add_kernel<<<(ROWELEMS + 255) / 256, 256, 0, stream>>>(x0, qpe, qc, ROWELEMS);
  gemm_kernel<Cc, 3 * Cc><<<rowTiles, 256, 0, stream>>>(qc, sa_wqkv, sa_bqkv, qkv, M, 0);
  counts_kernel<<<1, 256, 0, stream>>>(view, counts);
  {
    dim3 grid(NQn / 8, Hh, Bq);
    attn_kernel<<<grid, 256, 0, stream>>>(qkv, view, attnO);
  }
  gemm_kernel<Cc, Cc><<<rowTiles, 256, 0, stream>>>(attnO, sa_wo, sa_bo, tmp, M, 0);
  ln_kernel<<<(M + 7) / 8, 256, 0, stream>>>(tmp, x0, ln1_g, ln1_b, x1, M,
                                             nullptr, nullptr);
  // 6) deformable cross-attention branch
  add_kernel<<<(ROWELEMS + 255) / 256, 256, 0, stream>>>(x1, qpe, qc, ROWELEMS);
  gemm_kernel<Cc, 256><<<rowTiles, 256, 0, stream>>>(qc, d_off_w, d_off_b, offb, M, 0);
  gemm_kernel<Cc, Cc><<<rowTiles, 256, 0, stream>>>(qc, d_attn_w, d_attn_b, awb, M, 0);
  softmax16_kernel<<<(M * Hh + 255) / 256, 256, 0, stream>>>(awb, M * Hh);
  {
    dim3 grid(NQn, Bq);
    deform_kernel<<<grid, 128, 0, stream>>>(value, offb, awb, qpos, view, samp);
  }
  gemm_kernel<Cc, Cc><<<rowTiles, 256, 0, stream>>>(samp, d_out_w, d_out_b, tmp, M, 0);
  ln_kernel<<<(M + 7) / 8, 256, 0, stream>>>(tmp, x1, ln2_g, ln2_b, x2, M,
                                             nullptr, nullptr);
  // 7) FFN + ln3 (+ valid-view mask)
  gemm_kernel<Cc, FFNn><<<rowTiles, 256, 0, stream>>>(x2, ffn_w1, ffn_b1, ffnh, M, 1);
  gemm_kernel<FFNn, Cc><<<rowTiles, 256, 0, stream>>>(ffnh, ffn_w2, ffn_b2, tmp, M, 0);
  ln_kernel<<<(M + 7) / 8, 256, 0, stream>>>(tmp, x2, ln3_g, ln3_b, x3, M,
                                             view, counts);
  // 8) prediction head
  gemm_kernel<Cc, Cc><<<rowTiles, 256, 0, stream>>>(x3, ph_w1, ph_b1, phh, M, 1);
  gemm_kernel<Cc, 16><<<rowTiles, 256, 0, stream>>>(phh, w2pad, b2pad, res6, M, 0);
  // 9) outputs
  transpose_out_kernel<<<(ROWELEMS + 255) / 256, 256, 0, stream>>>(x3, o_x);
  final_kernel<<<(M + 255) / 256, 256, 0, stream>>>(res6, qpos, o_newpos,
                                                    o_center, o_offs, o_bbox);
}